// SpaAutoCorr_17076789969098
// MI455X (gfx1250) — compile-verified
//
#include <hip/hip_runtime.h>
#include <stdint.h>

// ---------------------------------------------------------------------------
// Moran's-I spatial autocorrelation MSE for MI455X (gfx1250).
//   numer[g] = sum_e v_e * (X[r_e,g]-m_g) * (X[c_e,g]-m_g)   (no SpMM temp)
// Memory-bound: 200MB stream + 6.4GB of (mostly L2-resident) row gathers.
// CDNA5 path: async global->LDS staging (ASYNCcnt), 4 edges per stage group,
// double buffered, so waits/barriers are amortized 4x and 32KB of async
// transfers are in flight while the current group computes.
// ---------------------------------------------------------------------------

#define K_CELLS 50000
#define K_GENES 512
#define K_EDGES 800000
#define K_EPS   1e-6f

#define EPG 4          // edges staged per group (4 x 8KB = 32KB per buffer)

// workspace layout (floats)
#define WS_SUMY 0
#define WS_SQY  512
#define WS_SUMP 1024
#define WS_SQP  1536
#define WS_NUMY 2048
#define WS_NUMP 2560
#define WS_W    3072
#define WS_TOT  3073

// ---- CDNA5 async global->LDS helpers --------------------------------------
__device__ __forceinline__ void asyncLoadB128ToLds(unsigned ldsByteAddr,
                                                   const float* gptr) {
  // vdst = LDS byte address VGPR, vaddr = 64-bit global address VGPR pair.
  asm volatile("global_load_async_to_lds_b128 %0, %1, off"
               :: "v"(ldsByteAddr), "v"(gptr)
               : "memory");
}

__device__ __forceinline__ void waitAsync0() {
#if __has_builtin(__builtin_amdgcn_s_wait_asynccnt)
  __builtin_amdgcn_s_wait_asynccnt(0);
#else
  asm volatile("s_wait_asynccnt 0" ::: "memory");
#endif
}

// ---- kernel 0: zero the accumulators --------------------------------------
__global__ void init_ws_kernel(float* __restrict__ ws) {
  int i = blockIdx.x * blockDim.x + threadIdx.x;
  if (i < WS_TOT) ws[i] = 0.0f;
}

// ---- kernel 1: per-gene sum and sum-of-squares (coalesced stream) ---------
__global__ __launch_bounds__(256)
void col_stats_kernel(const float* __restrict__ Y, const float* __restrict__ P,
                      float* __restrict__ ws, int rowsPerChunk) {
  const int g  = blockIdx.x * blockDim.x + threadIdx.x;   // gene id (0..511)
  const int r0 = blockIdx.y * rowsPerChunk;
  int r1 = r0 + rowsPerChunk;
  if (r1 > K_CELLS) r1 = K_CELLS;

  float sy = 0.f, sy2 = 0.f, sp = 0.f, sp2 = 0.f;
  for (int r = r0; r < r1; ++r) {
    float y = Y[(size_t)r * K_GENES + g];
    float p = P[(size_t)r * K_GENES + g];
    sy += y;  sy2 += y * y;
    sp += p;  sp2 += p * p;
  }
  atomicAdd(&ws[WS_SUMY + g], sy);
  atomicAdd(&ws[WS_SQY  + g], sy2);
  atomicAdd(&ws[WS_SUMP + g], sp);
  atomicAdd(&ws[WS_SQP  + g], sp2);
}

// ---- kernel 2: W = sum(vals) ----------------------------------------------
__global__ __launch_bounds__(256)
void wsum_kernel(const float* __restrict__ vals, float* __restrict__ ws) {
  __shared__ float red[256];
  const int gid    = blockIdx.x * blockDim.x + threadIdx.x;
  const int stride = gridDim.x * blockDim.x;
  float s = 0.f;
  for (int i = gid; i < K_EDGES; i += stride) s += vals[i];
  red[threadIdx.x] = s;
  __syncthreads();
  for (int off = 128; off > 0; off >>= 1) {
    if (threadIdx.x < off) red[threadIdx.x] += red[threadIdx.x + off];
    __syncthreads();
  }
  if (threadIdx.x == 0) atomicAdd(&ws[WS_W], red[0]);
}

// ---- kernel 3: edge accumulation, async double-buffered, 4 edges/group ----
// Block = 512 threads (16 wave32). Thread t owns gene t. Per staged edge the
// block moves 4 rows (Y[r], Y[c], P[r], P[c]) = 8KB: lane t moves one 16B
// chunk per edge with global_load_async_to_lds_b128 (512 lanes x 16B = 8KB).
__global__ __launch_bounds__(512)
void edge_kernel(const float* __restrict__ Y, const float* __restrict__ P,
                 const long long* __restrict__ rows,
                 const long long* __restrict__ cols,
                 const float* __restrict__ vals,
                 float* __restrict__ ws, int edgesPerBlock) {
  __shared__ float buf[2][EPG][4 * K_GENES];   // 2 x 32KB = 64KB

  const int t = threadIdx.x;
  const float invN = 1.0f / (float)K_CELLS;
  const float mY = ws[WS_SUMY + t] * invN;     // per-gene means (kernel 1 done)
  const float mP = ws[WS_SUMP + t] * invN;

  const int e0 = blockIdx.x * edgesPerBlock;
  int e1 = e0 + edgesPerBlock;
  if (e1 > K_EDGES) e1 = K_EDGES;
  if (e0 >= e1) return;
  const int nGroups = (e1 - e0 + EPG - 1) / EPG;

  // Which 16B chunk of which row this lane moves for every staged edge.
  const int which  = t >> 7;                   // 0:Y[r] 1:Y[c] 2:P[r] 3:P[c]
  const int within = t & 127;                  // 16B chunk index within row
  const float* srcBase = (which & 2) ? P : Y;
  const bool   useCol  = (which & 1) != 0;

  auto stage = [&](int b, int gIdx) {
    const int eBase = e0 + gIdx * EPG;
#pragma unroll
    for (int k = 0; k < EPG; ++k) {
      const int e = eBase + k;
      if (e < e1) {
        long long row = useCol ? cols[e] : rows[e];   // uniform scalar load
        const float* gp = srcBase + (size_t)row * K_GENES + (size_t)within * 4;
        const unsigned lds =
            (unsigned)(size_t)&buf[b][k][0] + (unsigned)t * 16u;
        asyncLoadB128ToLds(lds, gp);
      }
    }
  };

  float accY = 0.f, accP = 0.f;
  stage(0, 0);
  for (int gi = 0; gi < nGroups; ++gi) {
    waitAsync0();        // own wave's async transfers for group gi complete
    __syncthreads();     // ...and every other wave's too
    if (gi + 1 < nGroups) stage((gi + 1) & 1, gi + 1);   // prefetch next group

    const int eBase = e0 + gi * EPG;
#pragma unroll
    for (int k = 0; k < EPG; ++k) {
      const int e = eBase + k;
      if (e < e1) {
        const float* bb = buf[gi & 1][k];
        const float v  = vals[e];                        // uniform scalar load
        const float yr = bb[t],        yc = bb[512 + t];
        const float pr = bb[1024 + t], pc = bb[1536 + t];
        accY += v * (yr - mY) * (yc - mY);
        accP += v * (pr - mP) * (pc - mP);
      }
    }
    __syncthreads();     // all lanes done reading buf[gi&1] before re-staging
  }
  atomicAdd(&ws[WS_NUMY + t], accY);
  atomicAdd(&ws[WS_NUMP + t], accP);
}

// ---- kernel 4: finalize -> scalar MSE over genes --------------------------
__global__ __launch_bounds__(512)
void finalize_kernel(const float* __restrict__ ws, float* __restrict__ out) {
  __shared__ float red[512];
  const int g = threadIdx.x;
  const float Wv    = ws[WS_W];
  const float scale = (float)K_CELLS / Wv;
  const float invN  = 1.0f / (float)K_CELLS;

  float sy = ws[WS_SUMY + g];
  float dy = ws[WS_SQY + g] - sy * sy * invN;      // sum of centered^2
  if (dy == 0.0f) dy += K_EPS;
  float statY = scale * ws[WS_NUMY + g] / dy;

  float sp = ws[WS_SUMP + g];
  float dp = ws[WS_SQP + g] - sp * sp * invN;
  if (dp == 0.0f) dp += K_EPS;
  float statP = scale * ws[WS_NUMP + g] / dp;

  float d = statP - statY;
  red[g] = d * d;
  __syncthreads();
  for (int off = 256; off > 0; off >>= 1) {
    if (g < off) red[g] += red[g + off];
    __syncthreads();
  }
  if (g == 0) out[0] = red[0] * (1.0f / (float)K_GENES);
}

// ---------------------------------------------------------------------------
extern "C" void kernel_launch(void* const* d_in, const int* in_sizes, int n_in,
                              void* d_out, int out_size, void* d_ws, size_t ws_size,
                              hipStream_t stream) {
  const float*     Y    = (const float*)d_in[0];
  const float*     P    = (const float*)d_in[1];
  const long long* rows = (const long long*)d_in[2];   // int64 in reference
  const long long* cols = (const long long*)d_in[3];
  const float*     vals = (const float*)d_in[4];
  float* out = (float*)d_out;
  float* ws  = (float*)d_ws;
  (void)in_sizes; (void)n_in; (void)out_size; (void)ws_size;

  init_ws_kernel<<<(WS_TOT + 255) / 256, 256, 0, stream>>>(ws);

  const int rowsPerChunk = 512;
  dim3 gridStats(K_GENES / 256, (K_CELLS + rowsPerChunk - 1) / rowsPerChunk);
  col_stats_kernel<<<gridStats, 256, 0, stream>>>(Y, P, ws, rowsPerChunk);

  wsum_kernel<<<512, 256, 0, stream>>>(vals, ws);

  const int numEdgeBlocks = 2048;                       // ~391 edges/block
  const int epb = (K_EDGES + numEdgeBlocks - 1) / numEdgeBlocks;
  edge_kernel<<<numEdgeBlocks, 512, 0, stream>>>(Y, P, rows, cols, vals, ws, epb);

  finalize_kernel<<<1, 512, 0, stream>>>(ws, out);
}